// LRCoulomb_18580028522574
// MI455X (gfx1250) — compile-verified
//
#include <hip/hip_runtime.h>

typedef __attribute__((ext_vector_type(2))) float v2f;
typedef __attribute__((ext_vector_type(8))) float v8f;
typedef __attribute__((ext_vector_type(4))) int   v4i;

#define LRC_FACTOR 7.199822675975445f
#define LRC_RC     4.6f
#define LRC_INV_RC (1.0f / 4.6f)

__global__ void LRCoulomb_zero_kernel(float* __restrict__ out, int n) {
    int i = blockIdx.x * blockDim.x + threadIdx.x;
    if (i < n) out[i] = 0.0f;
}

// One wave32 per atom. Lane L loads 4 neighbor indices as one B128
// non-temporal load (512B coalesced per wave for the streamed nbmat),
// processes 4 pairs branchlessly, then the 32 per-lane partials are reduced
// entirely on the matrix pipe with two V_WMMA_F32_16X16X4_F32 ops
// (A = partials, B = ones). Lane 0 atomically adds the molecule contribution.
__global__ __launch_bounds__(256) void LRCoulomb_18580028522574_kernel(
    const float* __restrict__ coord,    // [(N+1)*3]
    const float* __restrict__ charges,  // [N+1]
    const int*   __restrict__ nbmat,    // [N*M]
    const int*   __restrict__ mol_idx,  // [N]
    float*       __restrict__ out,      // [N_MOL]
    int N, int M)
{
    const int wid  = threadIdx.x >> 5;
    const int lane = threadIdx.x & 31;
    const int atom = blockIdx.x * 8 + wid;
    if (atom >= N) return;  // wave-uniform guard: EXEC stays all-ones below

    const float qi  = charges[atom];
    const float rix = coord[3 * atom + 0];
    const float riy = coord[3 * atom + 1];
    const float riz = coord[3 * atom + 2];

    const v4i* __restrict__ row = (const v4i*)(nbmat + (size_t)atom * (size_t)M);
    const int nchunk = M >> 2;  // int4 chunks per row (M=128 -> 32, one per lane)

    float s = 0.0f;
    for (int c = lane; c < nchunk; c += 32) {
        // Streamed once -> non-temporal so it doesn't displace the hot
        // coord/charges gather targets in WGP$/L2.
        const v4i nb = __builtin_nontemporal_load(row + c);
#pragma unroll
        for (int k = 0; k < 4; ++k) {
            const int j = nb[k];
            // Gathers: coord (2.4MB) + charges (0.8MB) are L2-resident.
            const float dx = rix - coord[3 * j + 0];
            const float dy = riy - coord[3 * j + 1];
            const float dz = riz - coord[3 * j + 2];
            const float qj = charges[j];

            const float d2   = fmaf(dx, dx, fmaf(dy, dy, dz * dz));
            const float invd = __builtin_amdgcn_rsqf(d2);     // v_rsq_f32
            const float d    = d2 * invd;                     // = sqrt(d2)

            // Branchless smooth cutoff: exp(1 - 1/(1 - (d/rc)^2)), 0 outside.
            const float x     = d * LRC_INV_RC;
            const float inner = fmaxf(1.0f - x * x, 1e-12f);
            const float arg   = 1.0f - __builtin_amdgcn_rcpf(inner); // v_rcp_f32
            float fc          = __expf(arg);                  // v_exp_f32
            fc = (d < LRC_RC) ? fc : 0.0f;                    // v_cndmask

            const float e = (qi * qj) * invd * (1.0f - fc);
            const bool valid = (j != N) && (j != atom);       // pad & self mask
            s += valid ? e : 0.0f;
        }
    }

    // ---- 32-lane reduction entirely on the matrix pipe ----
    // WMMA #1: A(16x4) VGPR0 = s, VGPR1 = 0; B(4x16) = ones (layout-invariant).
    //   D[m][n] = s[m] + s[m+16]; rows 0-7 -> lanes 0-15, rows 8-15 -> lanes
    //   16-31, replicated across N. Per-lane sum of the 8 D VGPRs gives
    //   P = sum(rows 0-7) in lanes 0-15 and Q = sum(rows 8-15) in lanes 16-31.
    // WMMA #2: same trick on t -> D2[m][n] = t[m] + t[m+16] = P + Q = total,
    //   replicated in every element of every lane. No shuffle/DS op needed.
    v2f b; b.x = 1.0f; b.y = 1.0f;
    v8f cacc = {};

    v2f a1; a1.x = s; a1.y = 0.0f;
    v8f d1 = __builtin_amdgcn_wmma_f32_16x16x4_f32(
        false, a1, false, b, (short)0, cacc, false, false);

    float t = d1[0] + d1[1] + d1[2] + d1[3] + d1[4] + d1[5] + d1[6] + d1[7];

    v2f a2; a2.x = t; a2.y = 0.0f;
    v8f d2m = __builtin_amdgcn_wmma_f32_16x16x4_f32(
        false, a2, false, b, (short)0, cacc, false, false);

    const float total = d2m[0];

    if (lane == 0) {
        atomicAdd(&out[mol_idx[atom]], LRC_FACTOR * total);
    }
}

extern "C" void kernel_launch(void* const* d_in, const int* in_sizes, int n_in,
                              void* d_out, int out_size, void* d_ws, size_t ws_size,
                              hipStream_t stream) {
    (void)n_in; (void)d_ws; (void)ws_size;

    const float* coord   = (const float*)d_in[0];
    const float* charges = (const float*)d_in[1];
    const int*   nbmat   = (const int*)d_in[2];
    const int*   mol_idx = (const int*)d_in[3];
    float*       out     = (float*)d_out;

    const int N = in_sizes[0] / 3 - 1;      // coord is [(N+1)*3]
    const int M = in_sizes[2] / N;          // nbmat is [N*M]
    const int nmol = out_size;

    // Zero the accumulation bins (harness poisons d_out; we use atomics).
    LRCoulomb_zero_kernel<<<(nmol + 255) / 256, 256, 0, stream>>>(out, nmol);

    // 8 waves (atoms) per 256-thread block.
    const int blocks = (N + 7) / 8;
    LRCoulomb_18580028522574_kernel<<<blocks, 256, 0, stream>>>(
        coord, charges, nbmat, mol_idx, out, N, M);
}